// HELoss_20401094656627
// MI455X (gfx1250) — compile-verified
//
#include <hip/hip_runtime.h>
#include <hip/hip_bf16.h>

// ---------------------------------------------------------------------------
// HELoss on MI455X (gfx1250, wave32).
// logits: [B=4096, C=32000] f32, labels: [B] int32, cm: scalar f32.
// Memory-bound: 512 MB streamed once -> ~22 us floor at 23.3 TB/s.
// Structure: 1 block per row, coalesced non-temporal b128 loads, uniform
// branch-free inner loop, per-thread f32 partial sums, wave reduction via
// V_WMMA_F32_16X16X4_F32 (B = ones) + ds_swizzle SWAPX16, LDS combine,
// second 1-block kernel for the deterministic mean.  No atomics.
// ---------------------------------------------------------------------------

typedef __attribute__((ext_vector_type(2))) float v2f;
typedef __attribute__((ext_vector_type(4))) float v4f;
typedef __attribute__((ext_vector_type(8))) float v8f;

#define HE_S 15.0f
// 15 * log2(e) : exp(15x) == exp2(21.640425613334451 * x), one v_mul + v_exp_f32
#define HE_S_LOG2E 21.640425613334451f

__device__ __forceinline__ float expS(float x) {
  return __builtin_amdgcn_exp2f(HE_S_LOG2E * x);
}

// Full 32-lane sum via one f32 WMMA.
// A (16x4): lanes 0-15 hold {K0,K1} for M=lane; lanes 16-31 hold {K2,K3} for
// M=lane-16.  VGPR0 = partial, VGPR1 = 0 => A[m][0]=s_m, A[m][2]=s_{m+16}.
// B = all ones (4x16) => D[m][n] = s_m + s_{m+16} for every n.
// C/D (16x16 f32): lane n<16 VGPR r holds M=r; lane n>=16 holds M=r+8.
// Sum of the 8 C regs per lane gives a half-sum; SWAPX16 swizzle completes it.
// Requires EXEC all-1s: call only from uniform control flow.
__device__ __forceinline__ float wave_sum32_wmma(float s) {
  v2f a; a[0] = s;    a[1] = 0.0f;
  v2f b; b[0] = 1.0f; b[1] = 1.0f;
  v8f c = {};
  c = __builtin_amdgcn_wmma_f32_16x16x4_f32(
      /*neg_a=*/false, a, /*neg_b=*/false, b,
      /*c_mod=*/(short)0, c, /*reuse_a=*/false, /*reuse_b=*/false);
  float t = ((c[0] + c[1]) + (c[2] + c[3])) + ((c[4] + c[5]) + (c[6] + c[7]));
  // SWAPX16: group-of-32 swizzle, xor_mask=0x10, and_mask=0x1f -> 0x401F
  const int tswz = __builtin_amdgcn_ds_swizzle(__builtin_bit_cast(int, t), 0x401F);
  t += __builtin_bit_cast(float, tswz);
  return t;  // full wave sum in every lane
}

// 256-thread block reduction (8 waves). Result valid in thread 0.
__device__ __forceinline__ float block_sum256(float s) {
  __shared__ float wsum[8];
  const float w = wave_sum32_wmma(s);
  const int lane = threadIdx.x & 31;
  const int wid  = threadIdx.x >> 5;
  if (lane == 0) wsum[wid] = w;
  __syncthreads();
  float tot = 0.0f;
  if (threadIdx.x == 0) {
#pragma unroll
    for (int i = 0; i < 8; ++i) tot += wsum[i];
  }
  return tot;
}

// Branchless masked exp-sum of one v4f (columns col..col+3), excluding the
// target column.
__device__ __forceinline__ float exp4_masked(v4f x, int col, int label) {
  float e0 = expS(x.x); e0 = (col     == label) ? 0.0f : e0;
  float e1 = expS(x.y); e1 = (col + 1 == label) ? 0.0f : e1;
  float e2 = expS(x.z); e2 = (col + 2 == label) ? 0.0f : e2;
  float e3 = expS(x.w); e3 = (col + 3 == label) ? 0.0f : e3;
  return (e0 + e1) + (e2 + e3);
}

// One block per row: denominator partial sums + per-row loss term.
__global__ void heloss_row_kernel(const float* __restrict__ logits,
                                  const int* __restrict__ labels,
                                  const float* __restrict__ cm,
                                  float* __restrict__ row_out,
                                  int C) {
  const int row        = blockIdx.x;
  const long long base = (long long)row * (long long)C;
  const v4f* lp        = (const v4f*)(logits + base);
  const int label      = labels[row];
  const int nvec       = C >> 2;                       // 8000 (C multiple of 4)
  const int nfull      = nvec / (int)blockDim.x;       // 31 uniform iterations
  const int ntail      = nvec - nfull * (int)blockDim.x;  // 64 tail vectors

  float sum = 0.0f;
  int v = threadIdx.x;
#pragma unroll 4
  for (int it = 0; it < nfull; ++it, v += blockDim.x) {
    const v4f x = __builtin_nontemporal_load(&lp[v]);
    sum += exp4_masked(x, v << 2, label);
  }
  if ((int)threadIdx.x < ntail) {
    const int vt = nfull * (int)blockDim.x + (int)threadIdx.x;
    const v4f x = __builtin_nontemporal_load(&lp[vt]);
    sum += exp4_masked(x, vt << 2, label);
  }

  const float tot = block_sum256(sum);  // uniform: EXEC all-1s at the WMMA
  if (threadIdx.x == 0) {
    const float tgt   = logits[base + label];
    const float num   = HE_S * (tgt - cm[0]);
    const float denom = __expf(num) + tot;
    row_out[row] = num - __logf(denom);
  }
}

// Single block: deterministic mean over B per-row terms, negate.
__global__ void heloss_final_kernel(const float* __restrict__ row_terms,
                                    float* __restrict__ out,
                                    int B) {
  float s = 0.0f;
  for (int i = threadIdx.x; i < B; i += blockDim.x) s += row_terms[i];
  const float tot = block_sum256(s);
  if (threadIdx.x == 0) out[0] = -tot / (float)B;
}

extern "C" void kernel_launch(void* const* d_in, const int* in_sizes, int n_in,
                              void* d_out, int out_size, void* d_ws, size_t ws_size,
                              hipStream_t stream) {
  const float* logits = (const float*)d_in[0];
  const int*   labels = (const int*)d_in[1];   // int32 (JAX x64 disabled)
  const float* cm     = (const float*)d_in[2];
  float*       out    = (float*)d_out;
  float*       rowbuf = (float*)d_ws;          // B floats of scratch

  const int B = in_sizes[1];                   // 4096
  const int C = in_sizes[0] / B;               // 32000

  heloss_row_kernel<<<B, 256, 0, stream>>>(logits, labels, cm, rowbuf, C);
  heloss_final_kernel<<<1, 256, 0, stream>>>(rowbuf, out, B);
}